// ObjectLoss_85813446574803
// MI455X (gfx1250) — compile-verified
//
#include <hip/hip_runtime.h>
#include <hip/hip_bf16.h>
#include <math.h>

// ---------------------------------------------------------------------------
// ObjectLoss for MI455X (gfx1250, wave32).
// loss = mean over 6,451,200 logits of (1-y)x + (1 - y/2) softplus(-x)
//      = [ sum softplus(x) ]  +  [ sum over scattered cells of -y(x + 0.5*softplus(-x)) ]
// Pass A: streaming softplus reduction (HBM-bound, ~129MB @ 23.3 TB/s).
// Pass B: sparse ground-truth construction + correction (tiny).
// ---------------------------------------------------------------------------

typedef __attribute__((ext_vector_type(4))) float f4_t;
typedef __attribute__((ext_vector_type(2))) float v2f;
typedef __attribute__((ext_vector_type(8))) float v8f;

#define IOU_THR 0.5f
#define N_TOTAL 6451200.0

__device__ __forceinline__ float splus(float z) {
    // jax.nn.softplus(z) == max(z,0) + log1p(exp(-|z|))
    return fmaxf(z, 0.0f) + log1pf(expf(-fabsf(z)));
}

// ---------------------------------------------------------------------------
// Pass A: sum softplus(logit) over one det tensor.
// Layout (..., 5): logits at dword index 5k+4. We tile 20 dwords (4 logits)
// per thread per iteration: within the tile the logits live at f4 slots
// (+1).x (+2).y (+3).z (+4).w. Non-temporal B128 loads: single-use stream.
// Wave partials are reduced with V_WMMA_F32_16X16X4_F32 (A = partials 16x4,
// B = ones 4x16 -> D[m,n] = partial(lane m) + partial(lane m+16)).
// ---------------------------------------------------------------------------
__global__ void __launch_bounds__(256) obj_sum_kernel(const float* __restrict__ det,
                                                      unsigned int ngroups,
                                                      double* __restrict__ acc) {
    const f4_t* p = reinterpret_cast<const f4_t*>(det);
    unsigned int tid    = blockIdx.x * blockDim.x + threadIdx.x;
    unsigned int stride = gridDim.x * blockDim.x;

    float partial = 0.0f;
    for (unsigned int g = tid; g < ngroups; g += stride) {
        const f4_t* base = p + (size_t)g * 5u;   // 20 dwords per group
        f4_t a = __builtin_nontemporal_load(base + 1);
        f4_t b = __builtin_nontemporal_load(base + 2);
        f4_t c = __builtin_nontemporal_load(base + 3);
        f4_t d = __builtin_nontemporal_load(base + 4);
        partial += splus(a.x);
        partial += splus(b.y);
        partial += splus(c.z);
        partial += splus(d.w);
    }

    // ---- wave32 reduction via the matrix pipe (exact f32 math) ----
    // A (16x4 f32): lane L<16 holds row L {K0,K1}; lane L>=16 holds row L-16 {K2,K3}.
    // With a = {partial, 0} and B = all-ones: D[m,n] = partial(m) + partial(m+16).
    v2f amat; amat[0] = partial; amat[1] = 0.0f;
    v2f bmat; bmat[0] = 1.0f;    bmat[1] = 1.0f;
    v8f cacc = {0.f,0.f,0.f,0.f,0.f,0.f,0.f,0.f};
    v8f dres = __builtin_amdgcn_wmma_f32_16x16x4_f32(
        /*neg_a=*/false, amat, /*neg_b=*/false, bmat,
        /*c_mod=*/(short)0, cacc, /*reuse_a=*/false, /*reuse_b=*/false);
    // D VGPR i, lanes 0-15 = rowsum(i); lanes 16-31 = rowsum(i+8).
    float t = dres[0] + dres[1] + dres[2] + dres[3]
            + dres[4] + dres[5] + dres[6] + dres[7];
    // lanes 0-15: sum rows 0..7 ; lanes 16-31: sum rows 8..15. Wave sum = t(0)+t(16).

    __shared__ float s_w[16];                 // 8 waves * 2 halves
    unsigned int lane = threadIdx.x & 31u;
    unsigned int wv   = threadIdx.x >> 5;
    if (lane == 0)  s_w[wv * 2 + 0] = t;
    if (lane == 16) s_w[wv * 2 + 1] = t;
    __syncthreads();
    if (threadIdx.x == 0) {
        double bs = 0.0;
        int nw = (int)(blockDim.x >> 5);
        for (int i = 0; i < 2 * nw; ++i) bs += (double)s_w[i];
        atomicAdd(acc, bs);
    }
}

// ---------------------------------------------------------------------------
// Pass B: one block per batch b, thread t = target index.
// Recreates _calc_overlap + _suppress + scatter(set, drop) semantics, then
// adds the correction  -y * (x + 0.5*softplus(-x))  for each live cell.
// Scatter collisions within one (b, scale) grid: last target index wins.
// ---------------------------------------------------------------------------
__global__ void __launch_bounds__(64) obj_corr_kernel(const float* __restrict__ det0,
                                                      const float* __restrict__ det1,
                                                      const float* __restrict__ det2,
                                                      const float* __restrict__ anchors,
                                                      const float* __restrict__ targets,
                                                      double* __restrict__ acc) {
    const int b = blockIdx.x;
    const int t = threadIdx.x;

    __shared__ float  s_vals[60][3][3];
    __shared__ short  s_cy[60][3];
    __shared__ short  s_cx[60][3];
    __shared__ int    s_valid[60];
    __shared__ double s_part[64];

    const int SH[3] = {160, 80, 40};

    if (t < 60) {
        float tx = targets[((size_t)b * 60 + t) * 4 + 0];
        float ty = targets[((size_t)b * 60 + t) * 4 + 1];
        float tw = targets[((size_t)b * 60 + t) * 4 + 2];
        float th = targets[((size_t)b * 60 + t) * 4 + 3];
        int valid = (tx != -1.0f) && (ty != -1.0f) && (tw != -1.0f) && (th != -1.0f);
        s_valid[t] = valid;

        float ov[3][3];
        bool  hasnan = false;
        float mx = -__builtin_inff();
        for (int s = 0; s < 3; ++s) {
            float W = (float)SH[s];            // grids are square: h == w
            float zx = tx * W, zy = ty * W, zw = tw * W, zh = th * W;
            float xr = zx - (floorf(zx) + 0.5f);
            float yr = zy - (floorf(zy) + 0.5f);
            s_cy[t][s] = (short)(int)(ty * W);  // cells = (y*h, x*w).astype(int32)
            s_cx[t][s] = (short)(int)(tx * W);
            for (int a = 0; a < 3; ++a) {
                float ax = anchors[((size_t)s * 3 + a) * 2 + 0];
                float ay = anchors[((size_t)s * 3 + a) * 2 + 1];
                float lbx = fmaxf(-ax * 0.5f, xr - zw * 0.5f);
                float ubx = fminf( ax * 0.5f, xr + zw * 0.5f);
                float lby = fmaxf(-ay * 0.5f, yr - zh * 0.5f);
                float uby = fminf( ay * 0.5f, yr + zh * 0.5f);
                float m     = ((lbx < ubx) && (lby < uby)) ? 1.0f : 0.0f;
                float inter = (ubx - lbx) * (uby - lby) * m;
                float aarea = ax * ay;
                float tarea = (zw - xr) * (zh - yr);   // matches reference exactly
                float o = inter / (aarea + tarea - inter);
                ov[s][a] = o;
                if (o != o) hasnan = true;             // jnp.max propagates NaN
                else        mx = fmaxf(mx, o);
            }
        }
        // _suppress: max->1.0 ; <thr->0.0 ; (!=0 && !=1)->-1.0
        for (int s = 0; s < 3; ++s)
            for (int a = 0; a < 3; ++a) {
                float o = ov[s][a];
                float v;
                if (!hasnan && o == mx)   v =  1.0f;
                else if (o < IOU_THR)     v =  0.0f;   // NaN<thr is false
                else if (o == 1.0f)       v =  1.0f;   // survives step 3
                else                      v = -1.0f;   // includes NaN
                s_vals[t][s][a] = v;
            }
    }
    __syncthreads();

    double part = 0.0;
    if (t < 60 && s_valid[t]) {
        for (int s = 0; s < 3; ++s) {
            int cy = s_cy[t][s], cx = s_cx[t][s];
            bool win = true;                            // last writer wins
            for (int t2 = t + 1; t2 < 60; ++t2)
                if (s_valid[t2] && s_cy[t2][s] == cy && s_cx[t2][s] == cx) { win = false; break; }
            if (!win) continue;
            const float* det = (s == 0) ? det0 : (s == 1) ? det1 : det2;
            int W = SH[s];
            for (int a = 0; a < 3; ++a) {
                float y = s_vals[t][s][a];
                if (y != 0.0f) {
                    size_t idx = (((((size_t)b * 3 + a) * W + cy) * W + cx) * 5) + 4;
                    float x = det[idx];
                    part -= (double)y * ((double)x + 0.5 * (double)splus(-x));
                }
            }
        }
    }
    s_part[t] = part;
    __syncthreads();
    if (t == 0) {
        double bs = 0.0;
        for (int i = 0; i < 64; ++i) bs += s_part[i];
        atomicAdd(acc, bs);
    }
}

__global__ void obj_final_kernel(const double* __restrict__ acc, float* __restrict__ out) {
    if (blockIdx.x == 0 && threadIdx.x == 0)
        out[0] = (float)(acc[0] / N_TOTAL);
}

extern "C" void kernel_launch(void* const* d_in, const int* in_sizes, int n_in,
                              void* d_out, int out_size, void* d_ws, size_t ws_size,
                              hipStream_t stream) {
    (void)n_in; (void)out_size; (void)ws_size;
    const float* det0    = (const float*)d_in[0];
    const float* det1    = (const float*)d_in[1];
    const float* det2    = (const float*)d_in[2];
    const float* anchors = (const float*)d_in[3];
    const float* targets = (const float*)d_in[4];
    double* acc = (double*)d_ws;

    hipMemsetAsync(d_ws, 0, sizeof(double), stream);

    const unsigned int g0 = (unsigned int)(in_sizes[0] / 20);  // 1,228,800
    const unsigned int g1 = (unsigned int)(in_sizes[1] / 20);  //   307,200
    const unsigned int g2 = (unsigned int)(in_sizes[2] / 20);  //    76,800

    obj_sum_kernel<<<(g0 + 255) / 256, 256, 0, stream>>>(det0, g0, acc);
    obj_sum_kernel<<<(g1 + 255) / 256, 256, 0, stream>>>(det1, g1, acc);
    obj_sum_kernel<<<(g2 + 255) / 256, 256, 0, stream>>>(det2, g2, acc);

    obj_corr_kernel<<<64, 64, 0, stream>>>(det0, det1, det2, anchors, targets, acc);

    obj_final_kernel<<<1, 1, 0, stream>>>(acc, (float*)d_out);
}